// RuleAggregationLayer_44006234915594
// MI455X (gfx1250) — compile-verified
//
#include <hip/hip_runtime.h>

// Problem constants (match reference shapes: C=1, O=64, L=500, F=16; N from in_sizes).
constexpr int kO = 64;
constexpr int kL = 500;
constexpr int kF = 16;
constexpr int kBins = kL * kF;  // 8000 floats = 32 KB

typedef __attribute__((ext_vector_type(2))) float v2f;
typedef __attribute__((ext_vector_type(8))) float v8f;

// ---------------------------------------------------------------------------
// Kernel 0: zero the global bin accumulator in d_ws (must re-zero every call;
// the harness poisons d_ws once and never restores it between replays).
// ---------------------------------------------------------------------------
__global__ void zero_bins_kernel(float* __restrict__ gbins) {
    int i = blockIdx.x * blockDim.x + threadIdx.x;
    if (i < kBins) gbins[i] = 0.0f;
}

// ---------------------------------------------------------------------------
// Kernel 1: segment sum S[l,f] = sum_{v: label[v]==l} x[v,f].
// 4 lanes cooperate on one node: each lane loads one float4 (16B) of the
// node's 16-float feature row -> a wave32 issues 512B fully-coalesced loads.
// Accumulate into per-workgroup LDS bins with ds_add_f32, then flush each
// block's partial bins to the global accumulator with global_atomic_add_f32.
// Traffic: 32 MB (x) + labels ~ 1.5 us at 23.3 TB/s -- this is the roofline.
// 256 blocks x 1024 threads: enough WGPs in flight for full HBM bandwidth,
// while the per-block flush (256 x 8000 atomics ~ 8 MB) stays well under the
// x-stream traffic.
// ---------------------------------------------------------------------------
__global__ void __launch_bounds__(1024)
segsum_kernel(const float* __restrict__ x,
              const int* __restrict__ labels,
              float* __restrict__ gbins,
              int n) {
    __shared__ float lbins[kBins];  // 32 KB of the WGP's 320 KB LDS
    for (int j = threadIdx.x; j < kBins; j += blockDim.x) lbins[j] = 0.0f;
    __syncthreads();

    const float4* __restrict__ x4 = reinterpret_cast<const float4*>(x);
    const int total = n * 4;                       // one float4 per work item
    const int stride = gridDim.x * blockDim.x;
    for (int i = blockIdx.x * blockDim.x + threadIdx.x; i < total; i += stride) {
        // Prefetch next grid-stride iteration of the x stream (speculative;
        // OOB translations are silently dropped per CDNA5 prefetch rules).
        __builtin_prefetch(&x4[i + stride], 0, 0);

        const int v = i >> 2;        // node index
        const int q = i & 3;         // which float4 of the 16-float row
        const int lab = labels[v];
        const float4 xv = x4[i];
        const int base = lab * kF + q * 4;
        atomicAdd(&lbins[base + 0], xv.x);   // ds_add_f32
        atomicAdd(&lbins[base + 1], xv.y);
        atomicAdd(&lbins[base + 2], xv.z);
        atomicAdd(&lbins[base + 3], xv.w);
    }
    __syncthreads();

    for (int j = threadIdx.x; j < kBins; j += blockDim.x) {
        float p = lbins[j];
        atomicAdd(&gbins[j], p);             // global_atomic_add_f32
    }
}

// ---------------------------------------------------------------------------
// Kernel 2: out[o,f] = (1/N) * sum_l W[o,l] * S[l,f] + b[o,f]
// 64x500x16 f32 GEMM via V_WMMA_F32_16X16X4_F32.
// One block of 128 threads = 4 wave32s; wave w owns output rows [16w,16w+16).
// The 1/N scale is folded into the epilogue so the K-loop is pure
// loads + WMMA (no dependent VALU between load and matrix op).
// Per k-step of 4:
//   A (16x4 f32, 2 VGPRs): lanes 0-15 -> M=lane, K = k0+{0,1};
//                          lanes 16-31 -> M=lane-16, K = k0+{2,3}.
//   B (4x16 f32, 2 VGPRs): lanes 0-15 -> N=lane, K = k0+{0,1};
//                          lanes 16-31 -> N=lane-16, K = k0+{2,3}.
//   C/D (16x16 f32, 8 VGPRs): VGPR j -> M = j + (lane>=16 ? 8 : 0), N = lane&15.
// ---------------------------------------------------------------------------
__global__ void __launch_bounds__(128)
gemm_wmma_kernel(const float* __restrict__ W,     // (O, L) row-major
                 const float* __restrict__ S,     // (L, F) row-major
                 const float* __restrict__ bias,  // (O, F)
                 float* __restrict__ out,         // (O, F)
                 float invN) {
    const int lane = threadIdx.x & 31;
    const int wave = threadIdx.x >> 5;      // 0..3 -> M tile
    const int mbase = wave * 16;
    const int mn = lane & 15;               // M index for A, N index for B
    const int khalf = (lane >> 4) * 2;      // lanes 16-31 hold K+2,K+3

    const float* __restrict__ wrow = W + (mbase + mn) * kL;

    v8f acc = {};
    for (int l0 = 0; l0 < kL; l0 += 4) {
        const int k = l0 + khalf;
        v2f a, b;
        a.x = wrow[k];
        a.y = wrow[k + 1];
        b.x = S[k * kF + mn];
        b.y = S[(k + 1) * kF + mn];
        // 8 args: (neg_a, A, neg_b, B, c_mod, C, reuse_a, reuse_b)
        acc = __builtin_amdgcn_wmma_f32_16x16x4_f32(
            /*neg_a=*/false, a, /*neg_b=*/false, b,
            /*c_mod=*/(short)0, acc, /*reuse_a=*/false, /*reuse_b=*/false);
    }

    const int f = lane & 15;
    const int mhi = (lane >> 4) * 8;
#pragma unroll
    for (int j = 0; j < 8; ++j) {
        const int o = mbase + mhi + j;
        out[o * kF + f] = fmaf(acc[j], invN, bias[o * kF + f]);
    }
}

// ---------------------------------------------------------------------------
// Launch: zero bins -> segment sum -> WMMA GEMM (all stream-ordered).
// ---------------------------------------------------------------------------
extern "C" void kernel_launch(void* const* d_in, const int* in_sizes, int n_in,
                              void* d_out, int out_size, void* d_ws, size_t ws_size,
                              hipStream_t stream) {
    const float* x      = (const float*)d_in[0];  // (C, N, F) f32
    const float* W      = (const float*)d_in[1];  // (C*O*L,) f32
    const float* b      = (const float*)d_in[2];  // (C, O, F) f32
    const int*   labels = (const int*)d_in[3];    // (N,) int
    float* out   = (float*)d_out;                 // (C, O, F) f32
    float* gbins = (float*)d_ws;                  // (L, F) f32 scratch, 32 KB

    const int n = in_sizes[3];                    // N = 500000

    zero_bins_kernel<<<(kBins + 255) / 256, 256, 0, stream>>>(gbins);
    segsum_kernel<<<256, 1024, 0, stream>>>(x, labels, gbins, n);
    gemm_wmma_kernel<<<1, 128, 0, stream>>>(W, gbins, b, out, 1.0f / (float)n);
}